// LossClassInOut_31885837205884
// MI455X (gfx1250) — compile-verified
//
#include <hip/hip_runtime.h>

typedef float v2f __attribute__((ext_vector_type(2)));
typedef float v8f __attribute__((ext_vector_type(8)));

// workspace layout (floats)
#define WS_SUMS   0      // [2][32][32]
#define WS_CNT    2048   // [2][32]
#define WS_INTRA  2112   // [2][32]
#define WS_MEANS  2176   // [2][32][32]
#define WS_ACCUM_N 2176  // floats that must be zeroed each call

__global__ void k_zero(float* __restrict__ ws, int n) {
  for (int i = blockIdx.x * blockDim.x + threadIdx.x; i < n; i += gridDim.x * blockDim.x)
    ws[i] = 0.f;
}

// ---- DPP16 row-local (16-lane) reductions: no LDS traffic, VALU only ----
// stages: quad_perm(1,0,3,2)=0xB1, quad_perm(2,3,0,1)=0x4E,
//         row_half_mirror=0x141, row_mirror=0x140  (all row(16)-local)
template <int CTRL>
__device__ __forceinline__ float dpp_movf(float x) {
  return __int_as_float(__builtin_amdgcn_update_dpp(
      __float_as_int(x), __float_as_int(x), CTRL, 0xf, 0xf, true));
}
template <int CTRL>
__device__ __forceinline__ int dpp_movi(int x) {
  return __builtin_amdgcn_update_dpp(x, x, CTRL, 0xf, 0xf, true);
}
__device__ __forceinline__ float rowmax16(float v) {
  v = fmaxf(v, dpp_movf<0xB1>(v));
  v = fmaxf(v, dpp_movf<0x4E>(v));
  v = fmaxf(v, dpp_movf<0x141>(v));
  v = fmaxf(v, dpp_movf<0x140>(v));
  return v;
}
__device__ __forceinline__ int rowmin16i(int v) {
  v = min(v, dpp_movi<0xB1>(v));
  v = min(v, dpp_movi<0x4E>(v));
  v = min(v, dpp_movi<0x141>(v));
  v = min(v, dpp_movi<0x140>(v));
  return v;
}
__device__ __forceinline__ float rowsum16(float v) {
  v += dpp_movf<0xB1>(v);
  v += dpp_movf<0x4E>(v);
  v += dpp_movf<0x141>(v);
  v += dpp_movf<0x140>(v);
  return v;
}

// argmax over 32 columns of one row held as (vlo=cols 0..15, vhi=cols 16..31)
// across a 16-lane half. Exact first-index semantics (matches jnp.argmax).
__device__ __forceinline__ int argmax_pair(float vlo, float vhi, int col) {
  const float M = rowmax16(fmaxf(vlo, vhi));
  const int idx = (vlo == M) ? col : ((vhi == M) ? col + 16 : 64);
  return rowmin16i(idx);
}

// ---------------- pass 1: WMMA segment sums + class counts -----------------
__global__ void __launch_bounds__(256) k_pass1(
    const float* __restrict__ pred, const float* __restrict__ pseudo,
    float* __restrict__ ws, int N)
{
  __shared__ float lsum[2048];   // [2][32][32] block-level partial sums
  __shared__ float lcnt[64];     // [2][32] block-level counts
  for (int i = threadIdx.x; i < 2048; i += blockDim.x) lsum[i] = 0.f;
  if (threadIdx.x < 64) lcnt[threadIdx.x] = 0.f;
  __syncthreads();

  const int lane = threadIdx.x & 31;
  const int col  = lane & 15;
  const int half = lane >> 4;                       // 0: rows r,r+1 ; 1: rows r+2,r+3
  const int wpb  = blockDim.x >> 5;
  // wave-uniform loop bounds -> scalar loop control, EXEC stays all-ones at WMMA
  const int gwave  = __builtin_amdgcn_readfirstlane(blockIdx.x * wpb + (threadIdx.x >> 5));
  const int nWaves = __builtin_amdgcn_readfirstlane(gridDim.x * wpb);

  // D-tiles: a<P|Q><classHalf><dimHalf>, each 16x16 f32 accumulator
  v8f aP00 = {}, aP01 = {}, aP10 = {}, aP11 = {};
  v8f aQ00 = {}, aQ01 = {}, aQ10 = {}, aQ11 = {};
  // per-lane class counters: this half's two samples only; merged at flush.
  float caLo = 0.f, caHi = 0.f;   // pseudo-label counts for class col / col+16
  float cbLo = 0.f, cbHi = 0.f;   // pred-label  counts for class col / col+16

  const int nQuads = (N + 3) >> 2;
  for (int qd = gwave; qd < nQuads; qd += nWaves) {
    const int r  = qd << 2;
    const int s0 = r + 2 * half;        // K0 (lanes<16) / K2 (lanes>=16)
    const int s1 = s0 + 1;              // K1 / K3
    const bool ok0 = s0 < N, ok1 = s1 < N;
    const int o0 = (ok0 ? s0 : N - 1) * 32 + col;
    const int o1 = (ok1 ? s1 : N - 1) * 32 + col;
    const float p0l = pred[o0],   p0h = pred[o0 + 16];
    const float p1l = pred[o1],   p1h = pred[o1 + 16];
    const float q0l = pseudo[o0], q0h = pseudo[o0 + 16];
    const float q1l = pseudo[o1], q1h = pseudo[o1 + 16];

    const int qi0 = argmax_pair(q0l, q0h, col);   // pseudo labels
    const int qi1 = argmax_pair(q1l, q1h, col);
    const int pi0 = argmax_pair(p0l, p0h, col);   // pred labels
    const int pi1 = argmax_pair(p1l, p1h, col);

    // B tiles (4x16 f32): vgpr0 = K0/K2, vgpr1 = K1/K3
    const v2f Bp0 = {p0l, p1l}, Bp1 = {p0h, p1h};
    const v2f Bq0 = {q0l, q1l}, Bq1 = {q0h, q1h};
    // A tiles (16x4 f32 one-hot of labels): vgpr0 = K0/K2, vgpr1 = K1/K3
    const v2f Aq0 = { (ok0 && qi0 == col)      ? 1.f : 0.f, (ok1 && qi1 == col)      ? 1.f : 0.f };
    const v2f Aq1 = { (ok0 && qi0 == col + 16) ? 1.f : 0.f, (ok1 && qi1 == col + 16) ? 1.f : 0.f };
    const v2f Ap0 = { (ok0 && pi0 == col)      ? 1.f : 0.f, (ok1 && pi1 == col)      ? 1.f : 0.f };
    const v2f Ap1 = { (ok0 && pi0 == col + 16) ? 1.f : 0.f, (ok1 && pi1 == col + 16) ? 1.f : 0.f };

    // class counts fall out of the one-hot A tiles: no shuffles needed
    caLo += Aq0.x + Aq0.y;  caHi += Aq1.x + Aq1.y;
    cbLo += Ap0.x + Ap0.y;  cbHi += Ap1.x + Ap1.y;

    // pred segmented by pseudo labels; pseudo segmented by pred labels
    aP00 = __builtin_amdgcn_wmma_f32_16x16x4_f32(false, Aq0, false, Bp0, (short)0, aP00, false, false);
    aP01 = __builtin_amdgcn_wmma_f32_16x16x4_f32(false, Aq0, false, Bp1, (short)0, aP01, false, false);
    aP10 = __builtin_amdgcn_wmma_f32_16x16x4_f32(false, Aq1, false, Bp0, (short)0, aP10, false, false);
    aP11 = __builtin_amdgcn_wmma_f32_16x16x4_f32(false, Aq1, false, Bp1, (short)0, aP11, false, false);
    aQ00 = __builtin_amdgcn_wmma_f32_16x16x4_f32(false, Ap0, false, Bq0, (short)0, aQ00, false, false);
    aQ01 = __builtin_amdgcn_wmma_f32_16x16x4_f32(false, Ap0, false, Bq1, (short)0, aQ01, false, false);
    aQ10 = __builtin_amdgcn_wmma_f32_16x16x4_f32(false, Ap1, false, Bq0, (short)0, aQ10, false, false);
    aQ11 = __builtin_amdgcn_wmma_f32_16x16x4_f32(false, Ap1, false, Bq1, (short)0, aQ11, false, false);
  }

  // D layout: vgpr j -> class (j + 8*half) within tile, dim = col within tile
#pragma unroll
  for (int j = 0; j < 8; ++j) {
    const int c0 = j + 8 * half;
    atomicAdd(&lsum[(c0     ) * 32 + col     ], aP00[j]);
    atomicAdd(&lsum[(c0     ) * 32 + col + 16], aP01[j]);
    atomicAdd(&lsum[(c0 + 16) * 32 + col     ], aP10[j]);
    atomicAdd(&lsum[(c0 + 16) * 32 + col + 16], aP11[j]);
    atomicAdd(&lsum[1024 + (c0     ) * 32 + col     ], aQ00[j]);
    atomicAdd(&lsum[1024 + (c0     ) * 32 + col + 16], aQ01[j]);
    atomicAdd(&lsum[1024 + (c0 + 16) * 32 + col     ], aQ10[j]);
    atomicAdd(&lsum[1024 + (c0 + 16) * 32 + col + 16], aQ11[j]);
  }
  atomicAdd(&lcnt[col],      caLo);
  atomicAdd(&lcnt[col + 16], caHi);
  atomicAdd(&lcnt[32 + col], cbLo);
  atomicAdd(&lcnt[48 + col], cbHi);
  __syncthreads();
  for (int i = threadIdx.x; i < 2048; i += blockDim.x) atomicAdd(&ws[WS_SUMS + i], lsum[i]);
  if (threadIdx.x < 64) atomicAdd(&ws[WS_CNT + threadIdx.x], lcnt[threadIdx.x]);
}

// ---------------- means = sums / max(count,1) -----------------
__global__ void k_means(float* __restrict__ ws) {
  const int i = blockIdx.x * blockDim.x + threadIdx.x;
  if (i < 2048) {
    const int a = i >> 10, c = (i >> 5) & 31;
    ws[WS_MEANS + i] = ws[WS_SUMS + i] / fmaxf(ws[WS_CNT + a * 32 + c], 1.f);
  }
}

// ---------------- pass 2: intra-class L1 spread -----------------
__global__ void __launch_bounds__(256) k_pass2(
    const float* __restrict__ pred, const float* __restrict__ pseudo,
    float* __restrict__ ws, int N)
{
  __shared__ float sm[2048];  // means [2][32][32]
  __shared__ float bins[64];  // [2][32] intra accumulators
  for (int i = threadIdx.x; i < 2048; i += blockDim.x) sm[i] = ws[WS_MEANS + i];
  if (threadIdx.x < 64) bins[threadIdx.x] = 0.f;
  __syncthreads();

  const int lane = threadIdx.x & 31;
  const int col  = lane & 15;
  const int half = lane >> 4;
  const int wpb  = blockDim.x >> 5;
  const int gwave  = __builtin_amdgcn_readfirstlane(blockIdx.x * wpb + (threadIdx.x >> 5));
  const int nWaves = __builtin_amdgcn_readfirstlane(gridDim.x * wpb);

  const int nQuads = (N + 3) >> 2;
  for (int qd = gwave; qd < nQuads; qd += nWaves) {
    const int r  = qd << 2;
    const int s0 = r + 2 * half;
    const int s1 = s0 + 1;
    const bool ok0 = s0 < N, ok1 = s1 < N;
    const int o0 = (ok0 ? s0 : N - 1) * 32 + col;
    const int o1 = (ok1 ? s1 : N - 1) * 32 + col;
    const float p0l = pred[o0],   p0h = pred[o0 + 16];
    const float p1l = pred[o1],   p1h = pred[o1 + 16];
    const float q0l = pseudo[o0], q0h = pseudo[o0 + 16];
    const float q1l = pseudo[o1], q1h = pseudo[o1 + 16];

    const int qi0 = argmax_pair(q0l, q0h, col);
    const int qi1 = argmax_pair(q1l, q1h, col);
    const int pi0 = argmax_pair(p0l, p0h, col);
    const int pi1 = argmax_pair(p1l, p1h, col);

    float dp0 = fabsf(p0l - sm[qi0 * 32 + col]) + fabsf(p0h - sm[qi0 * 32 + col + 16]);
    float dp1 = fabsf(p1l - sm[qi1 * 32 + col]) + fabsf(p1h - sm[qi1 * 32 + col + 16]);
    float dq0 = fabsf(q0l - sm[1024 + pi0 * 32 + col]) + fabsf(q0h - sm[1024 + pi0 * 32 + col + 16]);
    float dq1 = fabsf(q1l - sm[1024 + pi1 * 32 + col]) + fabsf(q1h - sm[1024 + pi1 * 32 + col + 16]);
    dp0 = rowsum16(dp0);
    dp1 = rowsum16(dp1);
    dq0 = rowsum16(dq0);
    dq1 = rowsum16(dq1);
    if (col == 0) {  // lanes 0 (rows r,r+1) and 16 (rows r+2,r+3)
      if (ok0) { atomicAdd(&bins[qi0], dp0); atomicAdd(&bins[32 + pi0], dq0); }
      if (ok1) { atomicAdd(&bins[qi1], dp1); atomicAdd(&bins[32 + pi1], dq1); }
    }
  }
  __syncthreads();
  if (threadIdx.x < 64) atomicAdd(&ws[WS_INTRA + threadIdx.x], bins[threadIdx.x]);
}

// ---------------- finalize: scalar combine -----------------
__global__ void k_final(const float* __restrict__ ws, float* __restrict__ out, int N) {
  // 64 threads = 2 waves; wave a handles array a; lane = class c
  __shared__ float totals[64];   // [2][32] per-dim totals
  __shared__ float part[4];      // num/den per wave
  const int a = threadIdx.x >> 5;
  const int c = threadIdx.x & 31;
  const float* sums = ws + WS_SUMS + a * 1024;

  float t = 0.f;
  for (int cc = 0; cc < 32; ++cc) t += sums[cc * 32 + c];
  totals[a * 32 + c] = t;
  __syncthreads();

  const float cnt  = ws[WS_CNT + a * 32 + c];
  const float safe = fmaxf(cnt, 1.f);
  const float nr   = fmaxf((float)N - cnt, 1.f);
  float md = 0.f;
  for (int d = 0; d < 32; ++d) {
    const float s = sums[c * 32 + d];
    md += fabsf(s / safe - (totals[a * 32 + d] - s) / nr);
  }
  float diff = expf(-5.f * (md / 32.f));
  const bool present = cnt > 0.f;
  const int npres = __popcll(__ballot(present));
  if (npres <= 1) diff = 1.f;
  const float intra = ws[WS_INTRA + a * 32 + c] / (safe * 32.f);
  float numl = present ? (intra + diff) : 0.f;
  float denl = present ? 2.f : 0.f;
#pragma unroll
  for (int m = 1; m <= 16; m <<= 1) {
    numl += __shfl_xor(numl, m, 32);
    denl += __shfl_xor(denl, m, 32);
  }
  if (c == 0) { part[a * 2] = numl; part[a * 2 + 1] = denl; }
  __syncthreads();
  if (threadIdx.x == 0) out[0] = (part[0] + part[2]) / (part[1] + part[3]);
}

extern "C" void kernel_launch(void* const* d_in, const int* in_sizes, int n_in,
                              void* d_out, int out_size, void* d_ws, size_t ws_size,
                              hipStream_t stream) {
  const float* pred   = (const float*)d_in[0];
  const float* pseudo = (const float*)d_in[1];
  float* ws  = (float*)d_ws;
  float* out = (float*)d_out;
  const int N = in_sizes[0] / 32;

  const int nQuads = (N + 3) >> 2;
  int blocks = (nQuads + 7) / 8;          // 8 waves per 256-thread block
  if (blocks > 1024) blocks = 1024;
  if (blocks < 1) blocks = 1;

  k_zero <<<4, 256, 0, stream>>>(ws, WS_ACCUM_N);
  k_pass1<<<blocks, 256, 0, stream>>>(pred, pseudo, ws, N);
  k_means<<<8, 256, 0, stream>>>(ws);
  k_pass2<<<blocks, 256, 0, stream>>>(pred, pseudo, ws, N);
  k_final<<<1, 64, 0, stream>>>(ws, out, N);
}